// KKTLoss_31636729103247
// MI455X (gfx1250) — compile-verified
//
#include <hip/hip_runtime.h>

// KKT loss for batched sparse LPs on gfx1250 (MI455X).
// Pass 1: zero Ax/ATlam workspace (+ d_out).
// Pass 2: bandwidth-bound COO scatter: 128b NT streaming loads of
//         vals/rows/cols, cached gathers of x/lam, no-return f32 atomics
//         into 32KB-per-item accumulators (L2 resident).
// Pass 3: weighted sum-of-squares via V_WMMA_F32_16X16X4_F32 using the
//         D = A*A^T + C trick (diag(D) accumulates sums of squares).

#define B_ 64
#define M_ 8192
#define N_ 8192
#define NNZ_ 262144

typedef float v2f __attribute__((ext_vector_type(2)));
typedef float v4f __attribute__((ext_vector_type(4)));
typedef float v8f __attribute__((ext_vector_type(8)));
typedef int v4i __attribute__((ext_vector_type(4)));

// sqrt(weight / (B*M)) factors, folded into residuals so a single
// sum-of-squares accumulator chain computes the whole weighted loss.
#define S_PRIMAL 4.3673201e-04f  // sqrt(0.1 / 524288)
#define S_DUAL   4.3673201e-04f  // sqrt(0.1 / 524288)
#define S_STAT   1.0697706e-03f  // sqrt(0.6 / 524288)
#define S_COMP   6.1763235e-04f  // sqrt(0.2 / 524288)

__global__ __launch_bounds__(256) void kkt_zero_kernel(v4f* __restrict__ ws,
                                                       float* __restrict__ out) {
  const int t = blockIdx.x * 256 + threadIdx.x;  // 1024*256 = 262144 v4f = 1M floats
  v4f z = {0.f, 0.f, 0.f, 0.f};
  ws[t] = z;
  if (t == 0) out[0] = 0.f;
}

__global__ __launch_bounds__(256) void kkt_scatter_kernel(
    const v4f* __restrict__ vals4, const v4i* __restrict__ rows4,
    const v4i* __restrict__ cols4, const float* __restrict__ x,
    const float* __restrict__ lam, float* __restrict__ Ax,
    float* __restrict__ ATlam) {
  const int t = blockIdx.x * 256 + threadIdx.x;  // 0 .. B*NNZ/4-1
  const int b = t >> 16;                         // NNZ/4 = 65536 quads per item

  const float* __restrict__ xb = x + (size_t)b * N_;
  const float* __restrict__ lb = lam + (size_t)b * M_;
  float* __restrict__ axb = Ax + (size_t)b * M_;
  float* __restrict__ atb = ATlam + (size_t)b * N_;

  // Non-temporal 128-bit streaming loads: 192MB passes once, keep the hot
  // 8MB (x/lam/Ax/ATlam) resident in L2.
  const v4f v = __builtin_nontemporal_load(&vals4[t]);
  const v4i r = __builtin_nontemporal_load(&rows4[t]);
  const v4i c = __builtin_nontemporal_load(&cols4[t]);

  // A@x scatter into rows, A^T@lam scatter into cols (no-return f32 atomics).
  atomicAdd(&axb[r[0]], v[0] * xb[c[0]]);
  atomicAdd(&axb[r[1]], v[1] * xb[c[1]]);
  atomicAdd(&axb[r[2]], v[2] * xb[c[2]]);
  atomicAdd(&axb[r[3]], v[3] * xb[c[3]]);

  atomicAdd(&atb[c[0]], v[0] * lb[r[0]]);
  atomicAdd(&atb[c[1]], v[1] * lb[r[1]]);
  atomicAdd(&atb[c[2]], v[2] * lb[r[2]]);
  atomicAdd(&atb[c[3]], v[3] * lb[r[3]]);
}

__global__ __launch_bounds__(256) void kkt_reduce_kernel(
    const float* __restrict__ Ax, const float* __restrict__ ATlam,
    const float* __restrict__ lam, const float* __restrict__ bp,
    const float* __restrict__ cp, float* __restrict__ out) {
  const int t = blockIdx.x * 256 + threadIdx.x;  // grid 512 -> stride 131072
  v8f acc = {};                                  // C/D accumulator (16x16 f32)

  // 4 uniform iterations for every thread: EXEC is all-1s at each WMMA.
#pragma unroll
  for (int k = 0; k < 4; ++k) {
    const int i = t + k * (512 * 256);
    const float l = lam[i];
    const float axmb = Ax[i] - bp[i];
    const float atpc = ATlam[i] + cp[i];

    v2f a01, a23;
    a01[0] = fmaxf(axmb, 0.f) * S_PRIMAL;  // relu(Ax-b)
    a01[1] = fmaxf(-l, 0.f) * S_DUAL;      // relu(-lam)
    a23[0] = atpc * S_STAT;                // A^T lam + c
    a23[1] = (l * axmb) * S_COMP;          // lam * (Ax-b)

    // Same registers as A and B: the 16x4 A layout re-read as the 4x16 B
    // layout is exactly A^T, so D = A*A^T + C and diag(D) += sum(z^2).
    acc = __builtin_amdgcn_wmma_f32_16x16x4_f32(false, a01, false, a01,
                                                (short)0, acc, false, false);
    acc = __builtin_amdgcn_wmma_f32_16x16x4_f32(false, a23, false, a23,
                                                (short)0, acc, false, false);
  }

  // Extract diag(D). C layout: VGPR r, lanes 0-15 -> (M=r, N=lane);
  // lanes 16-31 -> (M=r+8, N=lane-16). So D[i][i] lives at:
  //   i in 0..7  : lane i,     acc[i]
  //   i in 8..15 : lane i+16,  acc[i-8]
  const int lane = threadIdx.x & 31;
  const int di = (lane < 8) ? lane : ((lane >= 24) ? (lane - 24) : -1);
  float sel = acc[0];
  sel = (di == 1) ? acc[1] : sel;
  sel = (di == 2) ? acc[2] : sel;
  sel = (di == 3) ? acc[3] : sel;
  sel = (di == 4) ? acc[4] : sel;
  sel = (di == 5) ? acc[5] : sel;
  sel = (di == 6) ? acc[6] : sel;
  sel = (di == 7) ? acc[7] : sel;
  float d = (di >= 0) ? sel : 0.f;

  // wave32 butterfly reduction
#pragma unroll
  for (int off = 16; off >= 1; off >>= 1) d += __shfl_xor(d, off, 32);

  __shared__ float ssum;
  if (threadIdx.x == 0) ssum = 0.f;
  __syncthreads();
  if (lane == 0) atomicAdd(&ssum, d);  // ds_add_f32, one per wave
  __syncthreads();
  if (threadIdx.x == 0) atomicAdd(out, ssum);  // 512 adds to the scalar
}

extern "C" void kernel_launch(void* const* d_in, const int* in_sizes, int n_in,
                              void* d_out, int out_size, void* d_ws, size_t ws_size,
                              hipStream_t stream) {
  const float* x_hat = (const float*)d_in[0];    // [B*N]
  const float* lam_hat = (const float*)d_in[1];  // [B*M]
  const float* A_vals = (const float*)d_in[2];   // [B, NNZ]
  const int* A_rows = (const int*)d_in[3];       // [B, NNZ]
  const int* A_cols = (const int*)d_in[4];       // [B, NNZ]
  const float* b_pad = (const float*)d_in[5];    // [B, M]
  const float* c_pad = (const float*)d_in[6];    // [B, N]
  float* out = (float*)d_out;

  float* Ax = (float*)d_ws;        // B*M floats (2 MB)
  float* ATlam = Ax + B_ * M_;     // B*N floats (2 MB)

  // Pass 1: zero 4 MB workspace + output scalar (re-done every call:
  // deterministic, survives the 0xAA poison and graph replays).
  kkt_zero_kernel<<<1024, 256, 0, stream>>>((v4f*)d_ws, out);

  // Pass 2: COO scatter, B*NNZ/4 = 4,194,304 threads, 4 nnz each.
  kkt_scatter_kernel<<<16384, 256, 0, stream>>>(
      (const v4f*)A_vals, (const v4i*)A_rows, (const v4i*)A_cols,
      x_hat, lam_hat, Ax, ATlam);

  // Pass 3: WMMA-based weighted sum-of-squares over B*M = 524288 elements.
  kkt_reduce_kernel<<<512, 256, 0, stream>>>(Ax, ATlam, lam_hat, b_pad, c_pad,
                                             out);
}